// GraphSAGENet_64295660421273
// MI455X (gfx1250) — compile-verified
//
#include <hip/hip_runtime.h>
#include <hip/hip_bf16.h>

typedef __attribute__((ext_vector_type(2))) float v2f;
typedef __attribute__((ext_vector_type(8))) float v8f;

#define EPS 1e-5f

// ---------------------------------------------------------------------------
// Kernel 1: zero scratch (deg + agg1 + agg2)
// ---------------------------------------------------------------------------
__global__ void k_zero(float* __restrict__ p, int n) {
    int i = blockIdx.x * blockDim.x + threadIdx.x;
    if (i < n) p[i] = 0.0f;
}

// ---------------------------------------------------------------------------
// Kernel 2: edge scatter for conv1. 16 lanes per edge: lanes 0..10 scatter the
// 11 features (coalesced 44B gather from x[src]), lane 11 counts degree.
// ---------------------------------------------------------------------------
__global__ void k_agg1(const long long* __restrict__ ei, const float* __restrict__ x,
                       float* __restrict__ agg1, float* __restrict__ deg, int E) {
    int t = blockIdx.x * blockDim.x + threadIdx.x;
    if (t >= E * 16) return;
    int e = t >> 4, f = t & 15;
    int dst = (int)ei[E + e];
    if (f < 11) {
        int src = (int)ei[e];
        atomicAdd(&agg1[dst * 11 + f], x[src * 11 + f]);
    } else if (f == 11) {
        atomicAdd(&deg[dst], 1.0f);
    }
}

// ---------------------------------------------------------------------------
// Kernel 3: conv1 per 16-node tile, one wave each.
//   out(16x32) = [agg1/deg | x](16x24pad) @ [w1_l ; w1_r](24x32) + b1_l
//   then BN1 + ReLU -> h1
// Uses V_WMMA_F32_16X16X4_F32 (full fp32 precision).
// ---------------------------------------------------------------------------
__global__ void k_conv1(const float* __restrict__ x, const float* __restrict__ agg1,
                        const float* __restrict__ deg,
                        const float* __restrict__ w_l, const float* __restrict__ b_l,
                        const float* __restrict__ w_r,
                        const float* __restrict__ g, const float* __restrict__ be,
                        const float* __restrict__ mu, const float* __restrict__ var,
                        float* __restrict__ h1, int ntiles) {
    int wave = (blockIdx.x * blockDim.x + threadIdx.x) >> 5;
    if (wave >= ntiles) return;            // wave-uniform branch: EXEC stays all-ones
    int lane = threadIdx.x & 31;
    int half = lane >> 4, r = lane & 15;
    int node = wave * 16 + r;
    float dinv = 1.0f / fmaxf(deg[node], 1.0f);

    auto feat = [&](int k) -> float {       // concat [mean_agg(11) | x(11) | pad]
        if (k < 11) return agg1[node * 11 + k] * dinv;
        if (k < 22) return x[node * 11 + (k - 11)];
        return 0.0f;
    };
    auto wc = [&](int k, int n) -> float {  // concat [w1_l ; w1_r ; 0]
        if (k < 11) return w_l[k * 32 + n];
        if (k < 22) return w_r[(k - 11) * 32 + n];
        return 0.0f;
    };

    v8f c0 = {}, c1 = {};
    #pragma unroll
    for (int kk = 0; kk < 6; ++kk) {        // K = 24 padded, 6 steps of 4
        int kb = kk * 4 + half * 2;
        v2f a;  a.x = feat(kb);      a.y = feat(kb + 1);
        v2f b0; b0.x = wc(kb, r);      b0.y = wc(kb + 1, r);
        v2f b1; b1.x = wc(kb, 16 + r); b1.y = wc(kb + 1, 16 + r);
        c0 = __builtin_amdgcn_wmma_f32_16x16x4_f32(false, a, false, b0, (short)0, c0, false, false);
        c1 = __builtin_amdgcn_wmma_f32_16x16x4_f32(false, a, false, b1, (short)0, c1, false, false);
    }

    #pragma unroll
    for (int nt = 0; nt < 2; ++nt) {
        v8f c = nt ? c1 : c0;
        int col = nt * 16 + r;
        float bb = b_l[col];
        float iv = rsqrtf(var[col] + EPS);
        float gg = g[col], bt = be[col], mm = mu[col];
        #pragma unroll
        for (int j = 0; j < 8; ++j) {
            int row = j + half * 8;
            float v = c[j] + bb;
            v = (v - mm) * iv * gg + bt;
            v = fmaxf(v, 0.0f);
            h1[(wave * 16 + row) * 32 + col] = v;
        }
    }
}

// ---------------------------------------------------------------------------
// Kernel 4: edge scatter for conv2. 32 lanes per edge, fully coalesced
// 128B gather from h1[src], 128B atomic scatter to agg2[dst].
// ---------------------------------------------------------------------------
__global__ void k_agg2(const long long* __restrict__ ei, const float* __restrict__ h1,
                       float* __restrict__ agg2, int E) {
    int t = blockIdx.x * blockDim.x + threadIdx.x;
    if (t >= E * 32) return;
    int e = t >> 5, f = t & 31;
    int src = (int)ei[e];
    int dst = (int)ei[E + e];
    atomicAdd(&agg2[dst * 32 + f], h1[src * 32 + f]);
}

// ---------------------------------------------------------------------------
// Kernel 5: conv2 + BN2 + ReLU + MLP(32->64->1), fused, one wave per 16 nodes.
// Grid is exactly tiles = N/16 waves (5 waves * 1250 blocks) so EXEC is all 1s.
// h2 tile staged through LDS to convert D-layout -> A-layout between WMMAs.
// ---------------------------------------------------------------------------
__global__ void __launch_bounds__(160) k_conv2_mlp(
        const float* __restrict__ h1, const float* __restrict__ agg2,
        const float* __restrict__ deg,
        const float* __restrict__ w_l, const float* __restrict__ b_l,
        const float* __restrict__ w_r,
        const float* __restrict__ g, const float* __restrict__ be,
        const float* __restrict__ mu, const float* __restrict__ var,
        const float* __restrict__ mw1, const float* __restrict__ mb1,
        const float* __restrict__ mw2, const float* __restrict__ mb2,
        float* __restrict__ out) {
    __shared__ float sh[5][16][33];        // +1 padding: conflict-free row/col access
    int wslot = threadIdx.x >> 5;
    int wave = blockIdx.x * 5 + wslot;
    int lane = threadIdx.x & 31;
    int half = lane >> 4, r = lane & 15;
    int node = wave * 16 + r;
    float dinv = 1.0f / fmaxf(deg[node], 1.0f);

    auto feat = [&](int k) -> float {       // concat [mean_agg(32) | h1(32)]
        if (k < 32) return agg2[node * 32 + k] * dinv;
        return h1[node * 32 + (k - 32)];
    };
    auto wc = [&](int k, int n) -> float {  // concat [w2_l ; w2_r]
        if (k < 32) return w_l[k * 32 + n];
        return w_r[(k - 32) * 32 + n];
    };

    // --- conv2: [16x64] @ [64x32] ---
    v8f c0 = {}, c1 = {};
    #pragma unroll
    for (int kk = 0; kk < 16; ++kk) {
        int kb = kk * 4 + half * 2;
        v2f a;  a.x = feat(kb);      a.y = feat(kb + 1);
        v2f b0; b0.x = wc(kb, r);      b0.y = wc(kb + 1, r);
        v2f b1; b1.x = wc(kb, 16 + r); b1.y = wc(kb + 1, 16 + r);
        c0 = __builtin_amdgcn_wmma_f32_16x16x4_f32(false, a, false, b0, (short)0, c0, false, false);
        c1 = __builtin_amdgcn_wmma_f32_16x16x4_f32(false, a, false, b1, (short)0, c1, false, false);
    }

    // bias + BN2 + ReLU, stage h2 tile to LDS (D-layout -> row-major)
    #pragma unroll
    for (int nt = 0; nt < 2; ++nt) {
        v8f c = nt ? c1 : c0;
        int col = nt * 16 + r;
        float bb = b_l[col];
        float iv = rsqrtf(var[col] + EPS);
        float gg = g[col], bt = be[col], mm = mu[col];
        #pragma unroll
        for (int j = 0; j < 8; ++j) {
            int row = j + half * 8;
            float v = c[j] + bb;
            v = (v - mm) * iv * gg + bt;
            sh[wslot][row][col] = fmaxf(v, 0.0f);
        }
    }
    // same-wave LDS store->load: DS ops are in-order per wave; wait for writes.
    asm volatile("s_wait_dscnt 0" ::: "memory");

    // --- MLP layer 1: h2[16x32] @ mlp_w1[32x64] ---
    v8f d[4] = {v8f{}, v8f{}, v8f{}, v8f{}};
    #pragma unroll
    for (int kk = 0; kk < 8; ++kk) {
        int kb = kk * 4 + half * 2;
        v2f a; a.x = sh[wslot][r][kb]; a.y = sh[wslot][r][kb + 1];
        #pragma unroll
        for (int nt = 0; nt < 4; ++nt) {
            int col = nt * 16 + r;
            v2f b; b.x = mw1[kb * 64 + col]; b.y = mw1[(kb + 1) * 64 + col];
            d[nt] = __builtin_amdgcn_wmma_f32_16x16x4_f32(false, a, false, b, (short)0, d[nt], false, false);
        }
    }

    // --- ReLU + fused 64->1 projection: per-row dot(h3, mlp_w2) ---
    float s[8];
    #pragma unroll
    for (int j = 0; j < 8; ++j) s[j] = 0.0f;
    #pragma unroll
    for (int nt = 0; nt < 4; ++nt) {
        int col = nt * 16 + r;
        float b1v = mb1[col];
        float w2v = mw2[col];
        #pragma unroll
        for (int j = 0; j < 8; ++j)
            s[j] += fmaxf(d[nt][j] + b1v, 0.0f) * w2v;
    }
    // butterfly sum over the 16 columns held by each 16-lane half (wave32)
    #pragma unroll
    for (int j = 0; j < 8; ++j) {
        #pragma unroll
        for (int m = 1; m <= 8; m <<= 1)
            s[j] += __shfl_xor(s[j], m, 32);
    }
    if (r == 0) {
        float bb = mb2[0];
        #pragma unroll
        for (int j = 0; j < 8; ++j)
            out[wave * 16 + j + half * 8] = s[j] + bb;   // lanes 0 & 16: rows j / j+8
    }
}

// ---------------------------------------------------------------------------
extern "C" void kernel_launch(void* const* d_in, const int* in_sizes, int n_in,
                              void* d_out, int out_size, void* d_ws, size_t ws_size,
                              hipStream_t stream) {
    const float*     x   = (const float*)d_in[0];
    const long long* ei  = (const long long*)d_in[1];  // int64 edge_index [2, E]
    const float* w1_l = (const float*)d_in[2];
    const float* b1_l = (const float*)d_in[3];
    const float* w1_r = (const float*)d_in[4];
    const float* bn1g = (const float*)d_in[5];
    const float* bn1b = (const float*)d_in[6];
    const float* bn1m = (const float*)d_in[7];
    const float* bn1v = (const float*)d_in[8];
    const float* w2_l = (const float*)d_in[9];
    const float* b2_l = (const float*)d_in[10];
    const float* w2_r = (const float*)d_in[11];
    const float* bn2g = (const float*)d_in[12];
    const float* bn2b = (const float*)d_in[13];
    const float* bn2m = (const float*)d_in[14];
    const float* bn2v = (const float*)d_in[15];
    const float* mw1  = (const float*)d_in[16];
    const float* mb1  = (const float*)d_in[17];
    const float* mw2  = (const float*)d_in[18];
    const float* mb2  = (const float*)d_in[19];
    float* out = (float*)d_out;

    const int N = in_sizes[0] / 11;         // 100000
    const int E = in_sizes[1] / 2;          // 3.2M
    const int tiles = N / 16;               // 6250

    float* ws   = (float*)d_ws;
    float* deg  = ws;                       // [N]
    float* agg1 = ws + N;                   // [N, 11]
    float* agg2 = ws + 12 * N;              // [N, 32]
    float* h1   = ws + 44 * N;              // [N, 32]

    int zn = 44 * N;                        // deg + agg1 + agg2
    k_zero<<<(zn + 255) / 256, 256, 0, stream>>>(ws, zn);

    k_agg1<<<(E * 16 + 255) / 256, 256, 0, stream>>>(ei, x, agg1, deg, E);

    k_conv1<<<(tiles + 7) / 8, 256, 0, stream>>>(x, agg1, deg, w1_l, b1_l, w1_r,
                                                 bn1g, bn1b, bn1m, bn1v, h1, tiles);

    k_agg2<<<(E * 32 + 255) / 256, 256, 0, stream>>>(ei, h1, agg2, E);

    k_conv2_mlp<<<tiles / 5, 160, 0, stream>>>(h1, agg2, deg, w2_l, b2_l, w2_r,
                                               bn2g, bn2b, bn2m, bn2v,
                                               mw1, mb1, mw2, mb2, out);
}